// MaskcomputeMoE2_56547539419490
// MI455X (gfx1250) — compile-verified
//
#include <hip/hip_runtime.h>

// ---------------------------------------------------------------------------
// Problem constants (from reference)
// ---------------------------------------------------------------------------
#define NUM_EXPERTS 64
#define HIDDEN      1024
#define TASK_NUM    8
#define BB          128
#define LL          512
#define CC          1024
#define W2_COLS     (2 * NUM_EXPERTS)          // 128
#define Y_ELEMS     ((size_t)BB * LL * CC)     // 67,108,864 floats
#define MASK_OFF    Y_ELEMS                    // expert_mask starts here (64)
#define TOK_OFF     (Y_ELEMS + NUM_EXPERTS)    // token_mask starts here (512)

typedef __attribute__((ext_vector_type(2))) float v2f;
typedef __attribute__((ext_vector_type(8))) float v8f;
typedef __attribute__((ext_vector_type(4))) float f4;

__device__ __forceinline__ float gelu_exact(float x) {
    return 0.5f * x * (1.0f + erff(x * 0.70710678118654752f));
}

// ---------------------------------------------------------------------------
// Kernel A: gating. One workgroup, 128 threads = 4 waves.
// Wave w computes logits[16w .. 16w+15] via V_WMMA_F32_16X16X4_F32:
//   D(16x16) += A(16x4) x B(4x16), A row 0 = hidden[kb..kb+3], rows 1..15 = 0.
// Then lane-0 thread does softmax(+1e-4) and top-2, publishes
// {g0, g1, e0, e1} to d_ws and writes expert_mask to d_out.
// ---------------------------------------------------------------------------
__global__ void __launch_bounds__(128)
moe_gating_wmma(const float* __restrict__ W1,
                const float* __restrict__ b1,
                const float* __restrict__ W2,
                const int*   __restrict__ task_index,
                float*       __restrict__ ws,
                float*       __restrict__ out)
{
    __shared__ float s_hid[HIDDEN];
    __shared__ float s_logits[NUM_EXPERTS];
    __shared__ int   s_e[2];

    const int tid = threadIdx.x;

    // Prefetch all of W2 (512 KB) into L2 so the WMMA k-loop hits cache.
    // Emits global_prefetch_b8 (CDNA5 prefetch path).
    for (int j = tid; j < (HIDDEN * W2_COLS * 4) / 128; j += 128) {
        __builtin_prefetch((const char*)W2 + (size_t)j * 128, 0, 3);
    }

    // hidden = gelu(W1[task] + b1), identical for every batch row.
    const int task = task_index[0];
    for (int h = tid; h < HIDDEN; h += 128) {
        s_hid[h] = gelu_exact(W1[task * HIDDEN + h] + b1[h]);
    }
    __syncthreads();

    // ---- WMMA matvec: logits = hidden @ W2[:, :64] ----
    const int ntile = tid >> 5;        // wave id = column tile (0..3)
    const int lane  = tid & 31;
    const int M     = lane & 15;       // row within tile
    const int hi2   = (lane >> 4) << 1; // lanes 0-15 -> K=0/1, lanes 16-31 -> K=2/3
    const float amask = (M == 0) ? 1.0f : 0.0f; // only A row 0 is the hidden vector
    const int col   = ntile * 16 + M;  // B column this lane supplies

    v8f acc = {};
    #pragma unroll 8
    for (int kb = 0; kb < HIDDEN; kb += 4) {
        const int r0 = kb + hi2;
        const float h0 = s_hid[r0];
        const float h1 = s_hid[r0 + 1];
        v2f a = { h0 * amask, h1 * amask };
        v2f b = { W2[(size_t)r0 * W2_COLS + col],
                  W2[(size_t)(r0 + 1) * W2_COLS + col] };
        // D = A x B + C  (16x16x4, full fp32)
        acc = __builtin_amdgcn_wmma_f32_16x16x4_f32(
                  false, a, false, b, (short)0, acc, false, false);
    }
    // D row 0 lives in acc[0] of lanes 0..15 (N = lane).
    if (lane < 16) s_logits[ntile * 16 + lane] = acc[0];
    __syncthreads();

    // ---- softmax + 1e-4, top-2 (tiny: 64 values, serial on thread 0) ----
    if (tid == 0) {
        float mx = -3.0e38f;
        for (int i = 0; i < NUM_EXPERTS; ++i) mx = fmaxf(mx, s_logits[i]);
        float pr[NUM_EXPERTS];
        float sum = 0.0f;
        for (int i = 0; i < NUM_EXPERTS; ++i) { pr[i] = __expf(s_logits[i] - mx); sum += pr[i]; }
        const float inv = 1.0f / sum;
        for (int i = 0; i < NUM_EXPERTS; ++i) pr[i] = pr[i] * inv + 1.0e-4f;
        int e0 = 0;
        for (int i = 1; i < NUM_EXPERTS; ++i) if (pr[i] > pr[e0]) e0 = i;
        int e1 = (e0 == 0) ? 1 : 0;
        for (int i = 0; i < NUM_EXPERTS; ++i)
            if (i != e0 && pr[i] > pr[e1]) e1 = i;
        ws[0] = pr[e0];               // g0
        ws[1] = pr[e1];               // g1
        ((int*)ws)[2] = e0;
        ((int*)ws)[3] = e1;
        s_e[0] = e0; s_e[1] = e1;
    }
    __syncthreads();

    // expert_mask: 128 at the two selected experts (all 128 batch rows pick them).
    if (tid < NUM_EXPERTS) {
        out[MASK_OFF + tid] =
            (tid == s_e[0] || tid == s_e[1]) ? (float)BB : 0.0f;
    }
}

// ---------------------------------------------------------------------------
// Kernel B: token_mask[l] = sum_b ( P[b,e0,l] + P[b,e1,l] ),  l in [0,512)
// ---------------------------------------------------------------------------
__global__ void __launch_bounds__(256)
moe_token_mask(const float* __restrict__ P,
               const float* __restrict__ ws,
               float*       __restrict__ out)
{
    const int* wsI = (const int*)ws;
    const int e0 = wsI[2];
    const int e1 = wsI[3];
    const int l = blockIdx.x * blockDim.x + threadIdx.x;
    if (l >= LL) return;
    float sum = 0.0f;
    for (int b = 0; b < BB; ++b) {
        const size_t base = (size_t)b * (NUM_EXPERTS * LL);
        sum += P[base + (size_t)e0 * LL + l] + P[base + (size_t)e1 * LL + l];
    }
    out[TOK_OFF + l] = sum;
}

// ---------------------------------------------------------------------------
// Kernel C: y[b,l,:] = input[b,l,:] * ( g0*P[b,e0,l] + g1*P[b,e1,l] )
// Bandwidth-bound streaming kernel: 512 MB total traffic -> ~22 us @ 23.3 TB/s.
// Non-temporal float4 load/store: every byte is touched exactly once, so keep
// it out of the 192 MB L2 working set.
// One block per (b,l) row: 256 threads x float4 = 4 KB row.
// ---------------------------------------------------------------------------
__global__ void __launch_bounds__(256)
moe_scale_rows(const float* __restrict__ in,
               const float* __restrict__ P,
               const float* __restrict__ ws,
               float*       __restrict__ out)
{
    const int* wsI = (const int*)ws;
    const float g0 = ws[0];
    const float g1 = ws[1];
    const int   e0 = wsI[2];
    const int   e1 = wsI[3];

    const int bid = blockIdx.x;         // 0 .. B*L-1
    const int b   = bid >> 9;           // / 512
    const int l   = bid & (LL - 1);

    const size_t pbase = (size_t)b * (NUM_EXPERTS * LL);
    const float s = g0 * P[pbase + (size_t)e0 * LL + l]
                  + g1 * P[pbase + (size_t)e1 * LL + l];

    const size_t off = (size_t)bid * CC + ((size_t)threadIdx.x << 2);
    f4 v = __builtin_nontemporal_load((const f4*)(in + off));
    v *= s;
    __builtin_nontemporal_store(v, (f4*)(out + off));
}

// ---------------------------------------------------------------------------
// Launch
// ---------------------------------------------------------------------------
extern "C" void kernel_launch(void* const* d_in, const int* in_sizes, int n_in,
                              void* d_out, int out_size, void* d_ws, size_t ws_size,
                              hipStream_t stream)
{
    const float* in_feat = (const float*)d_in[0];
    const float* P       = (const float*)d_in[1];
    const float* W1      = (const float*)d_in[2];
    const float* b1      = (const float*)d_in[3];
    const float* W2      = (const float*)d_in[4];
    const int*   task    = (const int*)  d_in[5];

    float* out = (float*)d_out;
    float* ws  = (float*)d_ws;

    // 1. Gating (WMMA matvec + softmax + top-2); publishes {g0,g1,e0,e1} to ws,
    //    writes expert_mask.
    moe_gating_wmma<<<1, 128, 0, stream>>>(W1, b1, W2, task, ws, out);

    // 2. token_mask (512 values).
    moe_token_mask<<<2, 256, 0, stream>>>(P, ws, out);

    // 3. y: bandwidth-bound streaming scale, one block per (b,l) row.
    moe_scale_rows<<<BB * LL, 256, 0, stream>>>(in_feat, P, ws, out);
}